// SagPooling_9852654977355
// MI455X (gfx1250) — compile-verified
//
#include <hip/hip_runtime.h>

// Problem constants (fixed by the reference)
#define NNODES 65536     // B * NPG
#define BGRAPH 64
#define NPG    1024
#define DIM    512
#define NEDGE  (NNODES * 32)
#define DROPK  512       // NPG - ceil(0.5*NPG)

typedef __attribute__((ext_vector_type(2))) float v2f;
typedef __attribute__((ext_vector_type(8))) float v8f;

// ---------------------------------------------------------------------------
// 1) in-degree: deg[dst[e]] += 1
// ---------------------------------------------------------------------------
__global__ void deg_kernel(const int* __restrict__ dst, float* __restrict__ deg, int E) {
    int e = blockIdx.x * blockDim.x + threadIdx.x;
    if (e < E) atomicAdd(&deg[dst[e]], 1.0f);
}

// ---------------------------------------------------------------------------
// 2) h[n] = dot(feat[n,:], weight) * norm[n]
//    16 nodes per wave via V_WMMA_F32_16X16X4_F32 accumulation over K=512.
//    A layout (16x4 f32): lanes 0-15 hold K=0,1; lanes 16-31 hold K=2,3.
//    B (4x16): column 0 = weight chunk -> only lanes 0 and 16 nonzero;
//    realized as an unconditional broadcast ds_load_b64 times a
//    loop-invariant 0/1 lane mask (no exec-mask divergence).
//    D column 0: lane 0 rows 0-7 (c[0..7]), lane 16 rows 8-15.
// ---------------------------------------------------------------------------
__global__ void __launch_bounds__(256)
matvec_kernel(const float* __restrict__ feat, const float* __restrict__ weight,
              const float* __restrict__ deg, float* __restrict__ h) {
    __shared__ __align__(16) float sw[DIM];
    for (int i = threadIdx.x; i < DIM; i += blockDim.x) sw[i] = weight[i];
    __syncthreads();

    const int  lane = threadIdx.x & 31;
    const int  wave = threadIdx.x >> 5;
    const int  tile = blockIdx.x * 8 + wave;     // 16 nodes per tile
    const int  m    = lane & 15;
    const bool hi   = lane >= 16;
    const int  koff = hi ? 2 : 0;
    const float sel = (m == 0) ? 1.0f : 0.0f;    // loop-invariant column-0 mask
    const float* frow = feat + (size_t)(tile * 16 + m) * DIM + koff;
    const float* wrow = sw + koff;

    v8f c = {};
#pragma unroll 4
    for (int k0 = 0; k0 < DIM; k0 += 4) {
        v2f a = *reinterpret_cast<const v2f*>(frow + k0);   // global_load_b64
        v2f wv = *reinterpret_cast<const v2f*>(wrow + k0);  // ds_load_b64 (broadcast)
        v2f b;
        b.x = wv.x * sel;
        b.y = wv.y * sel;
        c = __builtin_amdgcn_wmma_f32_16x16x4_f32(false, a, false, b,
                                                  (short)0, c, false, false);
    }

    if (m == 0) {
        const int base = tile * 16 + (hi ? 8 : 0);
#pragma unroll
        for (int r = 0; r < 8; ++r) {
            const int   node = base + r;
            const float d    = deg[node];
            const float nm   = (d > 0.0f) ? (1.0f / sqrtf(d)) : 0.0f;
            h[node] = c[r] * nm;
        }
    }
}

// ---------------------------------------------------------------------------
// 3) agg[dst[e]] += h[src[e]]
// ---------------------------------------------------------------------------
__global__ void edge_kernel(const int* __restrict__ src, const int* __restrict__ dst,
                            const float* __restrict__ h, float* __restrict__ agg, int E) {
    int e = blockIdx.x * blockDim.x + threadIdx.x;
    if (e < E) atomicAdd(&agg[dst[e]], h[src[e]]);
}

// ---------------------------------------------------------------------------
// 4) w[n] = relu(agg[n] * norm[n] + bias)
// ---------------------------------------------------------------------------
__global__ void score_kernel(const float* __restrict__ agg, const float* __restrict__ deg,
                             const float* __restrict__ bias, float* __restrict__ w) {
    int n = blockIdx.x * blockDim.x + threadIdx.x;
    if (n < NNODES) {
        float d  = deg[n];
        float nm = (d > 0.0f) ? (1.0f / sqrtf(d)) : 0.0f;
        w[n] = fmaxf(agg[n] * nm + bias[0], 0.0f);
    }
}

// ---------------------------------------------------------------------------
// 5) per-graph: zero the DROPK lowest-ranked scores (stable argsort rank,
//    exact tie handling; O(NPG) per thread over LDS broadcast reads)
// ---------------------------------------------------------------------------
__global__ void __launch_bounds__(NPG)
topk_kernel(float* __restrict__ w) {
    __shared__ float s[NPG];
    const int   g = blockIdx.x;
    const int   t = threadIdx.x;
    const float v = w[g * NPG + t];
    s[t] = v;
    __syncthreads();
    int rank = 0;
    for (int j = 0; j < NPG; ++j) {
        float u = s[j];
        rank += (u < v || (u == v && j < t)) ? 1 : 0;
    }
    if (rank < DROPK) w[g * NPG + t] = 0.0f;
}

// ---------------------------------------------------------------------------
// 6) out[n,d] = feat[n,d] * w[n]   (float4 vectorized; 256 MB traffic,
//    feat likely resident in 192MB L2 after matvec pass)
// ---------------------------------------------------------------------------
__global__ void gate_kernel(const float* __restrict__ feat, const float* __restrict__ wb,
                            float* __restrict__ out) {
    const int i = blockIdx.x * blockDim.x + threadIdx.x;   // float4 index
    const float s = wb[i >> 7];                            // 128 float4 per node
    const float4 f = reinterpret_cast<const float4*>(feat)[i];
    float4 o;
    o.x = f.x * s; o.y = f.y * s; o.z = f.z * s; o.w = f.w * s;
    reinterpret_cast<float4*>(out)[i] = o;
}

// ---------------------------------------------------------------------------
extern "C" void kernel_launch(void* const* d_in, const int* in_sizes, int n_in,
                              void* d_out, int out_size, void* d_ws, size_t ws_size,
                              hipStream_t stream) {
    (void)in_sizes; (void)n_in; (void)out_size; (void)ws_size;

    const float* feat   = (const float*)d_in[0];
    const int*   src    = (const int*)  d_in[1];
    const int*   dst    = (const int*)  d_in[2];
    const float* weight = (const float*)d_in[3];
    const float* bias   = (const float*)d_in[4];
    float*       out    = (float*)d_out;

    float* deg = (float*)d_ws;          // [NNODES]
    float* agg = deg + NNODES;          // [NNODES]
    float* h   = agg + NNODES;          // [NNODES]
    float* w   = h   + NNODES;          // [NNODES]

    // zero deg + agg (contiguous) — atomics accumulate into them every call
    hipMemsetAsync(d_ws, 0, (size_t)2 * NNODES * sizeof(float), stream);

    deg_kernel   <<<NEDGE / 256, 256, 0, stream>>>(dst, deg, NEDGE);
    matvec_kernel<<<NNODES / 128, 256, 0, stream>>>(feat, weight, deg, h);
    edge_kernel  <<<NEDGE / 256, 256, 0, stream>>>(src, dst, h, agg, NEDGE);
    score_kernel <<<NNODES / 256, 256, 0, stream>>>(agg, deg, bias, w);
    topk_kernel  <<<BGRAPH, NPG, 0, stream>>>(w);
    gate_kernel  <<<(NNODES * (DIM / 4)) / 256, 256, 0, stream>>>(feat, w, out);
}